// GCNHead_66383014527706
// MI455X (gfx1250) — compile-verified
//
#include <hip/hip_runtime.h>
#include <hip/hip_bf16.h>

typedef float v2f __attribute__((ext_vector_type(2)));
typedef float v8f __attribute__((ext_vector_type(8)));

#define DIM       256
#define NCLASSES  32
#define NGRAPHS   64

// ---------------------------------------------------------------------------
// K0: init deg (self-loop contributes 1), zero pooling scratch
// ---------------------------------------------------------------------------
__global__ void k_init(float* __restrict__ deg, float* __restrict__ sums,
                       float* __restrict__ cnt, int n_nodes) {
    int i = blockIdx.x * blockDim.x + threadIdx.x;
    if (i < n_nodes) deg[i] = 1.0f;
    if (i < NGRAPHS * NCLASSES) sums[i] = 0.0f;
    if (i < NGRAPHS) cnt[i] = 0.0f;
}

// ---------------------------------------------------------------------------
// K1: degree at destination (unit edge weights)
// ---------------------------------------------------------------------------
__global__ void k_deg(const int* __restrict__ dst, float* __restrict__ deg, int n_edges) {
    int e = blockIdx.x * blockDim.x + threadIdx.x;
    if (e < n_edges) atomicAdd(&deg[dst[e]], 1.0f);
}

// ---------------------------------------------------------------------------
// K2: dinv = rsqrt(deg) in place (deg >= 1 always due to self loops)
// ---------------------------------------------------------------------------
__global__ void k_rsqrt(float* __restrict__ deg, int n_nodes) {
    int i = blockIdx.x * blockDim.x + threadIdx.x;
    if (i < n_nodes) deg[i] = rsqrtf(deg[i]);
}

// ---------------------------------------------------------------------------
// K3: h = x @ W  via V_WMMA_F32_16X16X4_F32 (exact fp32 path).
// Block = 256 threads = 8 waves; each wave owns a 16x32 output tile.
// W (256x32 fp32, 32 KB) staged in LDS once per block.
// ---------------------------------------------------------------------------
__global__ void __launch_bounds__(256)
k_gemm(const float* __restrict__ x, const float* __restrict__ W,
       float* __restrict__ h, int n_nodes) {
    __shared__ float Ws[DIM * NCLASSES];  // 32 KB
    for (int i = threadIdx.x; i < DIM * NCLASSES; i += blockDim.x) Ws[i] = W[i];
    __syncthreads();

    const int wave    = threadIdx.x >> 5;
    const int lane    = threadIdx.x & 31;
    const int halfsel = lane >> 4;    // 0: lanes 0-15 (K pair 0,1); 1: lanes 16-31 (K pair 2,3)
    const int lrow    = lane & 15;

    const int row0 = (blockIdx.x * 8 + wave) * 16;
    // clamp address for out-of-grid waves; their stores are guarded off, and
    // EXEC stays all-1s around the WMMAs (ISA requirement).
    int arow = row0 + lrow;
    if (arow >= n_nodes) arow = 0;
    const float* xrow = x + (size_t)arow * DIM;

    v8f acc0 = {};  // columns 0..15
    v8f acc1 = {};  // columns 16..31

    #pragma unroll 4
    for (int k = 0; k < DIM; k += 4) {
        // stream-prefetch upcoming x data (emits global_prefetch_b8)
        __builtin_prefetch(xrow + k + 32, 0, 0);

        const int kk = k + halfsel * 2;
        // A fragment (16x4 fp32): lane<16 -> K=k,k+1 ; lane>=16 -> K=k+2,k+3
        v2f a;
        a.x = xrow[kk];
        a.y = xrow[kk + 1];
        // B fragments (4x16 fp32): VGPR0 = row K=kk, VGPR1 = row K=kk+1
        v2f b0, b1;
        b0.x = Ws[kk * NCLASSES + lrow];
        b0.y = Ws[(kk + 1) * NCLASSES + lrow];
        b1.x = Ws[kk * NCLASSES + 16 + lrow];
        b1.y = Ws[(kk + 1) * NCLASSES + 16 + lrow];

        acc0 = __builtin_amdgcn_wmma_f32_16x16x4_f32(false, a, false, b0,
                                                     (short)0, acc0, false, false);
        acc1 = __builtin_amdgcn_wmma_f32_16x16x4_f32(false, a, false, b1,
                                                     (short)0, acc1, false, false);
    }

    // C/D layout: VGPR r -> lanes 0-15: M=r, N=lane ; lanes 16-31: M=8+r, N=lane-16
    const int m_base = halfsel * 8;
    float* hp = h + (size_t)(row0 + m_base) * NCLASSES + lrow;
    if (row0 + 16 <= n_nodes) {
        // fast path: whole 16-row tile valid (always taken when N % 16 == 0)
        // -> straight-line, clause-friendly stores, no exec juggling
        #pragma unroll
        for (int r = 0; r < 8; ++r) {
            hp[(size_t)r * NCLASSES]      = acc0[r];
            hp[(size_t)r * NCLASSES + 16] = acc1[r];
        }
    } else if (row0 < n_nodes) {
        // ragged tail (unused for N=100000, kept for generality)
        #pragma unroll
        for (int r = 0; r < 8; ++r) {
            if (row0 + m_base + r < n_nodes) {
                hp[(size_t)r * NCLASSES]      = acc0[r];
                hp[(size_t)r * NCLASSES + 16] = acc1[r];
            }
        }
    }
}

// ---------------------------------------------------------------------------
// K4: accum[n][c] = h[n][c] * dinv[n]^2 + b[c]   (self-loop message + bias)
// ---------------------------------------------------------------------------
__global__ void k_accum_init(const float* __restrict__ h, const float* __restrict__ dinv,
                             const float* __restrict__ b, float* __restrict__ accum,
                             int n_nodes) {
    int i = blockIdx.x * blockDim.x + threadIdx.x;  // over N*32
    if (i < n_nodes * NCLASSES) {
        int n = i >> 5;
        int c = i & 31;
        float di = dinv[n];
        accum[i] = h[i] * di * di + b[c];
    }
}

// ---------------------------------------------------------------------------
// K5: edge gather-scale-scatter. One wave32 per edge, one lane per channel.
// h[src] gather and atomic adds to accum[dst] are fully coalesced per wave;
// h/accum (25 MB total) live in the 192 MB L2.
// ---------------------------------------------------------------------------
__global__ void __launch_bounds__(256)
k_edge(const int* __restrict__ src, const int* __restrict__ dst,
       const float* __restrict__ h, const float* __restrict__ dinv,
       float* __restrict__ accum, int n_edges) {
    const int lane = threadIdx.x & 31;
    const int e = blockIdx.x * (blockDim.x >> 5) + (threadIdx.x >> 5);
    if (e >= n_edges) return;
    const int s = src[e];
    const int d = dst[e];
    const float norm = dinv[s] * dinv[d];
    const float v = h[(size_t)s * NCLASSES + lane] * norm;
    atomicAdd(&accum[(size_t)d * NCLASSES + lane], v);
}

// ---------------------------------------------------------------------------
// K6: per-graph reduction. One wave32 per node, lane = channel.
// ---------------------------------------------------------------------------
__global__ void __launch_bounds__(256)
k_pool(const float* __restrict__ accum, const int* __restrict__ batch,
       float* __restrict__ sums, float* __restrict__ cnt, int n_nodes) {
    const int lane = threadIdx.x & 31;
    const int n = blockIdx.x * (blockDim.x >> 5) + (threadIdx.x >> 5);
    if (n >= n_nodes) return;
    const int g = batch[n];
    atomicAdd(&sums[g * NCLASSES + lane], accum[(size_t)n * NCLASSES + lane]);
    if (lane == 0) atomicAdd(&cnt[g], 1.0f);
}

// ---------------------------------------------------------------------------
// K7: out[g][c] = sums[g][c] / max(cnt[g], 1)
// ---------------------------------------------------------------------------
__global__ void k_final(const float* __restrict__ sums, const float* __restrict__ cnt,
                        float* __restrict__ out) {
    int i = blockIdx.x * blockDim.x + threadIdx.x;
    if (i < NGRAPHS * NCLASSES) {
        out[i] = sums[i] / fmaxf(cnt[i >> 5], 1.0f);
    }
}

// ---------------------------------------------------------------------------
extern "C" void kernel_launch(void* const* d_in, const int* in_sizes, int n_in,
                              void* d_out, int out_size, void* d_ws, size_t ws_size,
                              hipStream_t stream) {
    const float* x     = (const float*)d_in[0];
    const float* W     = (const float*)d_in[1];
    const float* b     = (const float*)d_in[2];
    const int*   ei    = (const int*)d_in[3];
    const int*   batch = (const int*)d_in[4];

    const int n_nodes = in_sizes[0] / DIM;
    const int n_edges = in_sizes[3] / 2;
    const int* src = ei;
    const int* dst = ei + n_edges;

    // workspace layout (floats): deg/dinv[N] | h[N*32] | accum[N*32] | sums[2048] | cnt[64]
    float* ws    = (float*)d_ws;
    float* deg   = ws;                              // reused as dinv in place
    float* h     = deg + n_nodes;
    float* accum = h + (size_t)n_nodes * NCLASSES;
    float* sums  = accum + (size_t)n_nodes * NCLASSES;
    float* cnt   = sums + NGRAPHS * NCLASSES;
    float* out   = (float*)d_out;

    const int B = 256;

    // K0: init
    k_init<<<(n_nodes + B - 1) / B, B, 0, stream>>>(deg, sums, cnt, n_nodes);
    // K1: degrees
    k_deg<<<(n_edges + B - 1) / B, B, 0, stream>>>(dst, deg, n_edges);
    // K2: dinv
    k_rsqrt<<<(n_nodes + B - 1) / B, B, 0, stream>>>(deg, n_nodes);
    // K3: WMMA GEMM (8 row-tiles of 16 per block)
    {
        int row_tiles = (n_nodes + 15) / 16;
        int blocks = (row_tiles + 7) / 8;
        k_gemm<<<blocks, B, 0, stream>>>(x, W, h, n_nodes);
    }
    // K4: self-loop + bias into accumulator
    k_accum_init<<<((n_nodes * NCLASSES) + B - 1) / B, B, 0, stream>>>(h, deg, b, accum, n_nodes);
    // K5: edge messages (wave per edge)
    k_edge<<<(n_edges + 7) / 8, B, 0, stream>>>(src, dst, h, deg, accum, n_edges);
    // K6: graph pooling (wave per node)
    k_pool<<<(n_nodes + 7) / 8, B, 0, stream>>>(accum, batch, sums, cnt, n_nodes);
    // K7: mean
    k_final<<<(NGRAPHS * NCLASSES + B - 1) / B, B, 0, stream>>>(sums, cnt, out);
}